// HDCTokenEncoder_67078799229486
// MI455X (gfx1250) — compile-verified
//
#include <hip/hip_runtime.h>
#include <math.h>

// Problem constants (B=8, S=2048, V=256, D=10000)
#define D_DIM 10000
#define S_DIM 2048
#define V_DIM 256
#define VEC4  2500          // D_DIM / 4 float4 chunks per row

// Typed, address-space-qualified pointers matching the async builtin signatures:
//   b128: (v4i AS1*, v4i AS3*, imm offset, imm cpol)
//   b32 : (int AS1*, int AS3*, imm offset, imm cpol)
typedef int v4i __attribute__((ext_vector_type(4)));
typedef __attribute__((address_space(1))) v4i* g4p;
typedef __attribute__((address_space(3))) v4i* l4p;
typedef __attribute__((address_space(1))) int* g1p;
typedef __attribute__((address_space(3))) int* l1p;

// Kernel 1: per-vocab-row inverse L2 norm (256 rows only; roll is a permutation
// so the norm of every output row equals the norm of its vocab row).
__global__ __launch_bounds__(256) void hdc_norms_kernel(const float* __restrict__ im,
                                                        float* __restrict__ invn) {
    __shared__ float red[256];
    const int v = blockIdx.x;
    const float* row = im + (size_t)v * D_DIM;
    float acc = 0.0f;
    for (int k = threadIdx.x; k < D_DIM; k += 256) {
        float x = row[k];
        acc += x * x;
    }
    red[threadIdx.x] = acc;
    __syncthreads();
    for (int s = 128; s > 0; s >>= 1) {
        if (threadIdx.x < s) red[threadIdx.x] += red[threadIdx.x + s];
        __syncthreads();
    }
    if (threadIdx.x == 0) {
        float n = sqrtf(red[0]);
        invn[v] = 1.0f / fmaxf(n, 1e-12f);
    }
}

// Kernel 2: one block per (b, i) row. Async-copy the vocab row into LDS
// (plus a 3-float cyclic tail so rolled float4 reads never wrap), then emit
// rolled, scaled float4 stores straight to HBM.
__global__ __launch_bounds__(256) void hdc_roll_kernel(const int* __restrict__ tok,
                                                       const float* __restrict__ im,
                                                       const float* __restrict__ invn,
                                                       float* __restrict__ out) {
    __shared__ float smem[D_DIM + 4];   // 40016 bytes
    const int r     = blockIdx.x;       // 0 .. B*S-1
    const int i     = r & (S_DIM - 1);  // roll amount (position in sequence)
    const int tid   = threadIdx.x;
    const int token = tok[r];           // block-uniform
    const float scale = invn[token];
    const float* rowp = im + (size_t)token * D_DIM;   // 16B aligned (40000 % 16 == 0)

    // Stage row -> LDS via gfx1250 async global->LDS path (ASYNCcnt).
    #pragma unroll
    for (int k = 0; k < 10; ++k) {
        int q = tid + k * 256;
        if (q < VEC4) {
            __builtin_amdgcn_global_load_async_to_lds_b128(
                (g4p)(rowp + 4 * q), (l4p)(&smem[4 * q]), 0, 0);
        }
    }
    // Cyclic tail: smem[D .. D+2] = row[0 .. 2]
    if (tid < 3) {
        __builtin_amdgcn_global_load_async_to_lds_b32(
            (g1p)(rowp + tid), (l1p)(&smem[D_DIM + tid]), 0, 0);
    }
    asm volatile("s_wait_asynccnt 0x0" ::: "memory");
    __syncthreads();

    const size_t base = (size_t)r * D_DIM;
    #pragma unroll
    for (int k = 0; k < 10; ++k) {
        int q = tid + k * 256;
        if (q < VEC4) {
            int d  = q * 4;
            int s0 = d - i;                 // d in [0,10000), i in [0,2048)
            if (s0 < 0) s0 += D_DIM;        // s0 in [0, D); s0+3 <= D+2 (tail covers)
            float4 val;
            val.x = smem[s0]     * scale;
            val.y = smem[s0 + 1] * scale;
            val.z = smem[s0 + 2] * scale;
            val.w = smem[s0 + 3] * scale;
            *reinterpret_cast<float4*>(out + base + d) = val;   // 16B aligned store
        }
    }
}

extern "C" void kernel_launch(void* const* d_in, const int* in_sizes, int n_in,
                              void* d_out, int out_size, void* d_ws, size_t ws_size,
                              hipStream_t stream) {
    const int*   tok  = (const int*)d_in[0];     // token_ids, B*S int32
    const float* im   = (const float*)d_in[1];   // item_memory, V*D float32
    float*       invn = (float*)d_ws;            // 256 floats of scratch
    float*       out  = (float*)d_out;           // B*S*D float32

    hdc_norms_kernel<<<V_DIM, 256, 0, stream>>>(im, invn);

    const int rows = in_sizes[0];                // B*S = 16384
    hdc_roll_kernel<<<rows, 256, 0, stream>>>(tok, im, invn, out);
}